// CorrectVQLoss_44263932953083
// MI455X (gfx1250) — compile-verified
//
#include <hip/hip_runtime.h>
#include <hip/hip_bf16.h>
#include <stdint.h>

// ---------------------------------------------------------------------------
// CorrectVQLoss for MI455X (gfx1250, wave32, WMMA).
//   B=16, C=512, T=1024, K=4096, N = B*T = 16384 rows.
// out[0] = mean((feats - codebook[code])^2)           (streaming, ~2us)
// out[1] = mean(argmin_k d2(feat_n, code_k) == code)  (68.7 GFLOP f32 GEMM,
//                                                      fused argmin, WMMA)
// Scratch layout (float* ws):  ws[0]=loss accum, ws[1]=match accum,
//                              ws[16..16+4095] = c2[k]   (needs 16.1 KB)
// ---------------------------------------------------------------------------

#define Bdim 16
#define Cdim 512
#define Tdim 1024
#define Kdim 4096
#define Nrows (Bdim * Tdim)          // 16384
#define A_STRIDE 514                 // 128-row A tile, padded (conflict-free)
#define B_STRIDE 130                 // 64-col B tile, padded
#define ROWS_PER_BLOCK 128
#define LDS_BYTES ((ROWS_PER_BLOCK * A_STRIDE + 64 * B_STRIDE) * 4)  // ~296 KB of 320 KB WGP LDS

typedef __attribute__((ext_vector_type(2))) float v2f;
typedef __attribute__((ext_vector_type(8))) float v8f;

// ---------------- init / finalize -----------------------------------------
__global__ void vq_init_kernel(float* ws) {
    if (threadIdx.x == 0) { ws[0] = 0.0f; ws[1] = 0.0f; }
}

__global__ void vq_finalize_kernel(const float* ws, float* out) {
    if (threadIdx.x == 0) {
        out[0] = ws[0] / (float)((long)Nrows * Cdim);   // mean squared error
        out[1] = ws[1] / (float)Nrows;                  // accuracy
    }
}

// ---------------- c2[k] = ||codebook[k]||^2  (block per row) ---------------
__global__ void vq_c2_kernel(const float* __restrict__ codebook, float* __restrict__ c2) {
    const int k = blockIdx.x;
    float s = 0.0f;
    for (int c = threadIdx.x; c < Cdim; c += 64) {
        float v = codebook[(size_t)k * Cdim + c];
        s += v * v;
    }
    #pragma unroll
    for (int off = 1; off < 32; off <<= 1) s += __shfl_xor(s, off, 32);
    __shared__ float red[2];
    if ((threadIdx.x & 31) == 0) red[threadIdx.x >> 5] = s;
    __syncthreads();
    if (threadIdx.x == 0) c2[k] = red[0] + red[1];
}

// ---------------- fused gather + MSE reduction -----------------------------
__global__ void vq_loss_kernel(const float* __restrict__ student,
                               const int*   __restrict__ codes,
                               const float* __restrict__ codebook,
                               float* __restrict__ ws) {
    const long total = (long)Bdim * Cdim * Tdim;      // 8388608
    float s = 0.0f;
    for (long e = blockIdx.x * (long)blockDim.x + threadIdx.x; e < total;
         e += (long)gridDim.x * blockDim.x) {
        int t = (int)(e & (Tdim - 1));
        int c = (int)((e >> 10) & (Cdim - 1));
        int b = (int)(e >> 19);
        float v = student[e];
        int   code = codes[(b << 10) + t];
        float d = v - codebook[(size_t)code * Cdim + c];
        s += d * d;
    }
    #pragma unroll
    for (int off = 1; off < 32; off <<= 1) s += __shfl_xor(s, off, 32);
    __shared__ float red[8];
    if ((threadIdx.x & 31) == 0) red[threadIdx.x >> 5] = s;
    __syncthreads();
    if (threadIdx.x == 0) {
        float t0 = 0.0f;
        #pragma unroll
        for (int w = 0; w < 8; ++w) t0 += red[w];
        atomicAdd(&ws[0], t0);
    }
}

// ---------------- main: GEMM (WMMA f32 16x16x4) + fused argmin -------------
// grid = 128 blocks, block = 256 threads (8 wave32), dynamic LDS = LDS_BYTES.
// Each wave owns 16 rows; all waves sweep the full codebook (64 groups of 64
// cols, C staged in 128-wide chunks), producing a 16x64 slab per group.
__global__ void vq_argmin_kernel(const float* __restrict__ student,
                                 const int*   __restrict__ codes,
                                 const float* __restrict__ codebook,
                                 const float* __restrict__ c2,
                                 float* __restrict__ ws_match) {
    extern __shared__ float smem[];
    float* As = smem;                                   // [128][A_STRIDE]
    float* Bs = smem + ROWS_PER_BLOCK * A_STRIDE;       // [64][B_STRIDE]

    const int tid  = threadIdx.x;
    const int lane = tid & 31;
    const int wave = tid >> 5;              // 0..7
    const int m    = lane & 15;             // row/col index within 16
    const int h    = lane >> 4;             // half select (K-pair / M-half)
    const int wrowL = wave * 16;            // wave's local row base

    // --- stage A tile: student[b, :, t0:t0+128] -> As[t_local][c] ---
    const int b  = blockIdx.x >> 3;                 // 1024/128 = 8 blocks per b
    const int t0 = (blockIdx.x & 7) * ROWS_PER_BLOCK;
    for (int i = tid; i < ROWS_PER_BLOCK * Cdim; i += 256) {
        int tl = i & (ROWS_PER_BLOCK - 1);
        int c  = i >> 7;
        As[tl * A_STRIDE + c] = student[((size_t)b * Cdim + c) * Tdim + t0 + tl];
    }

    float minv[8]; int mini[8];
    #pragma unroll
    for (int j = 0; j < 8; ++j) { minv[j] = 3.0e38f; mini[j] = 0; }

    for (int g = 0; g < Kdim / 64; ++g) {           // 64 column groups
        v8f acc[4] = {};                            // 16x64 slab per wave
        for (int ch = 0; ch < 4; ++ch) {            // C chunks of 128
            __syncthreads();
            // stage B: codebook[g*64 .. +63][ch*128 .. +127] (coalesced)
            for (int i = tid; i < 64 * 128; i += 256) {
                int col = i >> 7, cc = i & 127;
                Bs[col * B_STRIDE + cc] =
                    codebook[(size_t)(g * 64 + col) * Cdim + ch * 128 + cc];
            }
            // prefetch next chunk's codebook lines into cache
            if (ch < 3)
                __builtin_prefetch(&codebook[(size_t)(g * 64 + (tid >> 2)) * Cdim +
                                             (ch + 1) * 128 + (tid & 3) * 32], 0, 0);
            __syncthreads();

            const float* Ap  = As + (wrowL + m) * A_STRIDE + ch * 128 + 2 * h;
            const float* Bp  = Bs + m * B_STRIDE + 2 * h;
            #pragma unroll
            for (int s = 0; s < 32; ++s) {          // K-steps of 4 along C
                v2f a  = *reinterpret_cast<const v2f*>(Ap + 4 * s);
                v2f b0 = *reinterpret_cast<const v2f*>(Bp + 4 * s + 0 * 16 * B_STRIDE);
                v2f b1 = *reinterpret_cast<const v2f*>(Bp + 4 * s + 1 * 16 * B_STRIDE);
                v2f b2 = *reinterpret_cast<const v2f*>(Bp + 4 * s + 2 * 16 * B_STRIDE);
                v2f b3 = *reinterpret_cast<const v2f*>(Bp + 4 * s + 3 * 16 * B_STRIDE);
                acc[0] = __builtin_amdgcn_wmma_f32_16x16x4_f32(false, a, false, b0, (short)0, acc[0], false, false);
                acc[1] = __builtin_amdgcn_wmma_f32_16x16x4_f32(false, a, false, b1, (short)0, acc[1], false, false);
                acc[2] = __builtin_amdgcn_wmma_f32_16x16x4_f32(false, a, false, b2, (short)0, acc[2], false, false);
                acc[3] = __builtin_amdgcn_wmma_f32_16x16x4_f32(false, a, false, b3, (short)0, acc[3], false, false);
            }
        }
        // epilogue: d2 = c2[k] - 2*dot  (f2[n] constant per row -> dropped)
        #pragma unroll
        for (int t = 0; t < 4; ++t) {
            int coln = g * 64 + t * 16 + m;         // this lane's column
            float c2v = c2[coln];
            #pragma unroll
            for (int j = 0; j < 8; ++j) {           // C/D: VGPR j = row j+8h
                float d = fmaf(-2.0f, acc[t][j], c2v);
                if (d < minv[j]) { minv[j] = d; mini[j] = coln; }
            }
        }
    }

    // cross-lane argmin over the 16 lanes of each half (rows j+8h)
    #pragma unroll
    for (int off = 1; off < 16; off <<= 1) {
        #pragma unroll
        for (int j = 0; j < 8; ++j) {
            float ov = __shfl_xor(minv[j], off, 32);
            int   oi = __shfl_xor(mini[j], off, 32);
            if (ov < minv[j] || (ov == minv[j] && oi < mini[j])) {  // first-min tie-break
                minv[j] = ov; mini[j] = oi;
            }
        }
    }
    if (m == 0) {                                   // lanes 0 and 16
        int cnt = 0;
        int rowbase = blockIdx.x * ROWS_PER_BLOCK + wrowL + 8 * h;
        #pragma unroll
        for (int j = 0; j < 8; ++j)
            cnt += (mini[j] == codes[rowbase + j]) ? 1 : 0;
        atomicAdd(ws_match, (float)cnt);
    }
}

// ---------------------------------------------------------------------------
extern "C" void kernel_launch(void* const* d_in, const int* in_sizes, int n_in,
                              void* d_out, int out_size, void* d_ws, size_t ws_size,
                              hipStream_t stream) {
    const float* student  = (const float*)d_in[0];   // (16,512,1024) f32
    const int*   codes    = (const int*)  d_in[1];   // (1,16,1024) i32
    const float* codebook = (const float*)d_in[2];   // (4096,512) f32
    float* out = (float*)d_out;                      // [loss, accuracy]
    float* ws  = (float*)d_ws;                       // accums + c2
    float* c2  = ws + 16;

    (void)in_sizes; (void)n_in; (void)out_size; (void)ws_size;

    // Host-side, non-stream, deterministic config call (same every launch):
    // opt-in to >64KB dynamic LDS for the big A tile (296 KB of 320 KB WGP LDS).
    hipFuncSetAttribute((const void*)vq_argmin_kernel,
                        hipFuncAttributeMaxDynamicSharedMemorySize, LDS_BYTES);

    vq_init_kernel<<<1, 32, 0, stream>>>(ws);
    vq_c2_kernel<<<Kdim, 64, 0, stream>>>(codebook, c2);
    vq_loss_kernel<<<1024, 256, 0, stream>>>(student, codes, codebook, ws);
    vq_argmin_kernel<<<Nrows / ROWS_PER_BLOCK, 256, LDS_BYTES, stream>>>(
        student, codes, codebook, c2, &ws[1]);
    vq_finalize_kernel<<<1, 32, 0, stream>>>(ws, out);
}